// HomogeneousSharedEncoder_41652592837487
// MI455X (gfx1250) — compile-verified
//
#include <hip/hip_runtime.h>
#include <cstddef>

#define H 128

typedef float v2f __attribute__((ext_vector_type(2)));
typedef float v8f __attribute__((ext_vector_type(8)));

// ---------------------------------------------------------------- zero fill
__global__ void zero_u32_kernel(unsigned* __restrict__ p, long long n) {
    long long i = (long long)blockIdx.x * blockDim.x + threadIdx.x;
    if (i < n) p[i] = 0u;
}

// ------------------------------------------------- input projection (small K)
// out[i, :] = x[i, :K] @ W[K,128] + b   (128 threads = 128 output columns)
__global__ void proj_kernel(const float* __restrict__ x,
                            const float* __restrict__ W,
                            const float* __restrict__ b,
                            float* __restrict__ out, int n, int K) {
    __shared__ __align__(16) float sW[32 * H];
    const int tid = threadIdx.x;               // 0..127
    for (int i = tid; i < K * H; i += 128) sW[i] = W[i];
    const float bias = b[tid];
    __syncthreads();
    const int start = blockIdx.x * 64;
    const int end   = min(start + 64, n);
    for (int i = start; i < end; ++i) {
        const float* xr = x + (size_t)i * K;   // uniform address -> scalar loads
        float acc = bias;
        for (int k = 0; k < K; ++k) acc += xr[k] * sW[k * H + tid];
        out[(size_t)i * H + tid] = acc;
    }
}

// -------------------------------------------------------------- edge counts
__global__ void count_kernel(const int* __restrict__ tgt, int* __restrict__ cnt, int E) {
    int e = blockIdx.x * blockDim.x + threadIdx.x;
    if (e < E) atomicAdd(&cnt[tgt[e]], 1);
}

__global__ void inv_count_kernel(const int* __restrict__ cnt, float* __restrict__ inv, int n) {
    int i = blockIdx.x * blockDim.x + threadIdx.x;
    if (i < n) inv[i] = 1.0f / (float)max(cnt[i], 1);
}

// ----------------------------------------------- scatter-mean (pre-weighted)
// agg[tgt, c] += feat[src, c] * inv[tgt]   (mean via precomputed 1/max(cnt,1))
__global__ void scatter_kernel(const float* __restrict__ feat,
                               const int* __restrict__ src,
                               const int* __restrict__ tgt,
                               const float* __restrict__ inv,
                               float* __restrict__ agg, int E) {
    long long idx = (long long)blockIdx.x * blockDim.x + threadIdx.x;
    if (idx >= (long long)E * H) return;
    const int e = (int)(idx >> 7);
    const int c = (int)(idx & (H - 1));
    const int s = src[e];
    const int t = tgt[e];
    atomicAdd(&agg[(size_t)t * H + c], feat[(size_t)s * H + c] * inv[t]);
}

// ------------------------------------------------------- fused layer update
// out = elu(A0 @ W0 + A1 @ W1 + bias)  with fp32 WMMA (16x16x4).
// 8 waves/block, wave w owns rows [blk*128 + 16w, +16), all 8 column tiles.
// Weight matrix is staged into LDS in a pair-swizzled layout so each lane's
// B fragment {B[ka][col], B[ka+1][col]} is one aligned ds_load_b64:
//   sw[(k>>1)*256 + 2*col + (k&1)] = B[k][col]
__global__ __launch_bounds__(256) void fused_gemm_elu_kernel(
        const float* __restrict__ A0, const float* __restrict__ A1,
        const float* __restrict__ W0, const float* __restrict__ W1,
        const float* __restrict__ bias, float* __restrict__ out, int n) {
    __shared__ __align__(16) float sB[H * H];          // 64 KB (swizzled)
    const int tid  = threadIdx.x;
    const int wave = tid >> 5;
    const int lane = tid & 31;
    const int half = lane >> 4;                        // 0: lanes 0-15, 1: lanes 16-31
    const int l15  = lane & 15;
    const int m0   = blockIdx.x * 128 + wave * 16;

    int arow = m0 + l15;                               // clamp keeps EXEC all-ones
    if (arow >= n) arow = n - 1;
    const float* aRow0 = A0 + (size_t)arow * H;
    const float* aRow1 = A1 + (size_t)arow * H;

    v8f acc[8] = {};                                   // 8 column tiles of 16x16

    for (int phase = 0; phase < 2; ++phase) {
        __syncthreads();
        {   // stage 128x128 weights into LDS, pair-swizzled for b64 fragments
            const float4* s = (const float4*)(phase ? W1 : W0);
            for (int i = tid; i < H * H / 4; i += 256) {
                const float4 v = s[i];
                const int k  = i >> 5;                 // row (K index)
                const int c0 = (i & 31) << 2;          // first of 4 columns
                float* d = sB + ((k >> 1) << 8) + (k & 1);
                d[2 * (c0 + 0)] = v.x;
                d[2 * (c0 + 1)] = v.y;
                d[2 * (c0 + 2)] = v.z;
                d[2 * (c0 + 3)] = v.w;
            }
        }
        __syncthreads();
        const float* ap = phase ? aRow1 : aRow0;
        // per-lane base into swizzled LDS: pair row (k0>>1)+half, element 2*col
        const float* bbase = sB + (half << 8) + (l15 << 1);
        for (int k0 = 0; k0 < H; k0 += 4) {
            const int ka = k0 + (half << 1);           // A 16x4 fp32 lane layout
            v2f a;
            a.x = ap[ka];
            a.y = ap[ka + 1];
            const float* bk = bbase + ((k0 >> 1) << 8);
            #pragma unroll
            for (int t = 0; t < 8; ++t) {
                const v2f b = *(const v2f*)(bk + (t << 5));  // ds_load_b64
                acc[t] = __builtin_amdgcn_wmma_f32_16x16x4_f32(
                    false, a, false, b, (short)0, acc[t], false, false);
            }
        }
    }

    // epilogue: bias + ELU + guarded store (C/D layout: VGPR r -> M = r + 8*half)
    #pragma unroll
    for (int t = 0; t < 8; ++t) {
        const int col = (t << 4) + l15;
        const float bc = bias[col];
        #pragma unroll
        for (int r = 0; r < 8; ++r) {
            const int row = m0 + (half << 3) + r;
            if (row < n) {
                float v = acc[t][r] + bc;
                out[(size_t)row * H + col] = v > 0.0f ? v : (__expf(v) - 1.0f);
            }
        }
    }
}

// ---------------------------------------------------------------- launcher
extern "C" void kernel_launch(void* const* d_in, const int* in_sizes, int n_in,
                              void* d_out, int out_size, void* d_ws, size_t ws_size,
                              hipStream_t stream) {
    const float* x_int  = (const float*)d_in[0];
    const float* x_lane = (const float*)d_in[1];
    const float* x_sens = (const float*)d_in[2];
    const float* x_inj  = (const float*)d_in[3];
    const int*   e_sp   = (const int*)d_in[4];
    const int*   e_fl   = (const int*)d_in[5];
    const int*   e_fs   = (const int*)d_in[6];
    const int*   e_in   = (const int*)d_in[7];
    const float* W_int  = (const float*)d_in[8];
    const float* b_int  = (const float*)d_in[9];
    const float* W_lane = (const float*)d_in[10];
    const float* b_lane = (const float*)d_in[11];
    const float* W_sens = (const float*)d_in[12];
    const float* b_sens = (const float*)d_in[13];
    const float* W_inj  = (const float*)d_in[14];
    const float* b_inj  = (const float*)d_in[15];
    const float* self_W = (const float*)d_in[16];
    const float* self_b = (const float*)d_in[17];
    const float* rel_W  = (const float*)d_in[18];

    const int nInt  = in_sizes[0] / 32;
    const int nLane = in_sizes[1] / 16;
    const int nSens = in_sizes[2] / 8;
    const int nInj  = in_sizes[3] / 8;
    const int eSp   = in_sizes[4] / 2;
    const int eFl   = in_sizes[5] / 2;
    const int eFs   = in_sizes[6] / 2;
    const int eIn   = in_sizes[7] / 2;
    const int L     = in_sizes[16] / (H * H);

    // ---- workspace layout (floats) ----
    float* ws = (float*)d_ws;
    size_t o = 0;
    float* hA    = ws + o; o += (size_t)nInt  * H;   // ping
    float* hB    = ws + o; o += (size_t)nInt  * H;   // pong
    float* hLane = ws + o; o += (size_t)nLane * H;
    float* hSens = ws + o; o += (size_t)nSens * H;
    float* hInj  = ws + o; o += (size_t)nInj  * H;
    float* aggS  = ws + o; o += (size_t)nInt  * H;   // layer-invariant relation mean
    float* aggB  = ws + o; o += (size_t)nInt  * H;   // per-layer working aggregate
    int*   cnt   = (int*)(ws + o); o += (size_t)4 * nInt;
    float* invw  = ws + o; o += (size_t)4 * nInt;
    (void)ws_size; (void)n_in; (void)out_size;

    // ---- input projections ----
    proj_kernel<<<(nInt  + 63) / 64, 128, 0, stream>>>(x_int,  W_int,  b_int,  hA,    nInt,  32);
    proj_kernel<<<(nLane + 63) / 64, 128, 0, stream>>>(x_lane, W_lane, b_lane, hLane, nLane, 16);
    proj_kernel<<<(nSens + 63) / 64, 128, 0, stream>>>(x_sens, W_sens, b_sens, hSens, nSens, 8);
    proj_kernel<<<(nInj  + 63) / 64, 128, 0, stream>>>(x_inj,  W_inj,  b_inj,  hInj,  nInj,  8);

    // ---- per-relation in-degree -> 1/max(cnt,1) ----
    const long long zc = 4LL * nInt;
    zero_u32_kernel<<<(unsigned)((zc + 255) / 256), 256, 0, stream>>>((unsigned*)cnt, zc);
    count_kernel<<<(eSp + 255) / 256, 256, 0, stream>>>(e_sp + eSp, cnt + 0 * nInt, eSp);
    count_kernel<<<(eFl + 255) / 256, 256, 0, stream>>>(e_fl + eFl, cnt + 1 * nInt, eFl);
    count_kernel<<<(eFs + 255) / 256, 256, 0, stream>>>(e_fs + eFs, cnt + 2 * nInt, eFs);
    count_kernel<<<(eIn + 255) / 256, 256, 0, stream>>>(e_in + eIn, cnt + 3 * nInt, eIn);
    inv_count_kernel<<<(int)((zc + 255) / 256), 256, 0, stream>>>(cnt, invw, (int)zc);

    // ---- layer-invariant aggregate: means from lane / sens / inj sources ----
    const long long za = (long long)nInt * H;
    zero_u32_kernel<<<(unsigned)((za + 255) / 256), 256, 0, stream>>>((unsigned*)aggS, za);
    scatter_kernel<<<(unsigned)(((long long)eFl * H + 255) / 256), 256, 0, stream>>>(
        hLane, e_fl, e_fl + eFl, invw + 1 * nInt, aggS, eFl);
    scatter_kernel<<<(unsigned)(((long long)eFs * H + 255) / 256), 256, 0, stream>>>(
        hSens, e_fs, e_fs + eFs, invw + 2 * nInt, aggS, eFs);
    scatter_kernel<<<(unsigned)(((long long)eIn * H + 255) / 256), 256, 0, stream>>>(
        hInj,  e_in, e_in + eIn, invw + 3 * nInt, aggS, eIn);

    // ---- layers ----
    float* hcur = hA;
    float* hnext = hB;
    for (int l = 0; l < L; ++l) {
        hipMemcpyAsync(aggB, aggS, (size_t)nInt * H * sizeof(float),
                       hipMemcpyDeviceToDevice, stream);
        scatter_kernel<<<(unsigned)(((long long)eSp * H + 255) / 256), 256, 0, stream>>>(
            hcur, e_sp, e_sp + eSp, invw, aggB, eSp);
        float* outp = (l == L - 1) ? (float*)d_out : hnext;
        fused_gemm_elu_kernel<<<(nInt + 127) / 128, 256, 0, stream>>>(
            hcur, aggB,
            self_W + (size_t)l * H * H, rel_W + (size_t)l * H * H,
            self_b + (size_t)l * H, outp, nInt);
        float* t = hnext; hnext = hcur; hcur = t;
    }
}